// GraphTransformerWithYAtomMapPosEmbInefficientChargesSeparate_9612136808544
// MI455X (gfx1250) — compile-verified
//
#include <hip/hip_runtime.h>
#include <stdint.h>

#define NB 32
#define NN 512
#define NHALF 256

typedef unsigned int u32;
typedef u32 v4u __attribute__((ext_vector_type(4)));
typedef int  v4i __attribute__((ext_vector_type(4)));
typedef int  v8i __attribute__((ext_vector_type(8)));

#if defined(__has_builtin)
#  if __has_builtin(__builtin_amdgcn_tensor_load_to_lds) && __has_builtin(__builtin_amdgcn_s_wait_tensorcnt)
#    define USE_TDM 1
#  else
#    define USE_TDM 0
#  endif
#else
#  define USE_TDM 0
#endif

// Flat float offsets into d_out (X_cat, C_cat, E1, E2, BD1 in return order)
constexpr long long XCAT_OFF = 0;
constexpr long long CCAT_OFF = (long long)NB * NN * 32;
constexpr long long E1_OFF   = CCAT_OFF + (long long)NB * NN * 6;
constexpr long long E2_OFF   = E1_OFF + (long long)NB * NN * NN * 5;
constexpr long long BD1_OFF  = E2_OFF + (long long)NB * NN * NN * 5;

// ---------------------------------------------------------------------------
// Kernel 1: per-batch alignment map.  pp[i,j] = product position matching the
// atom-map number of reactant position j, or -1 if unmapped.
// ---------------------------------------------------------------------------
__global__ __launch_bounds__(NN) void build_pp_kernel(const int* __restrict__ amn,
                                                      const int* __restrict__ mol,
                                                      int* __restrict__ pp) {
  const int i = blockIdx.x;
  const int j = threadIdx.x;
  __shared__ int table[NN + 1];
  __shared__ int smax;
  const int a  = amn[i * NN + j];
  const int as = mol[i * NN + j];
  table[j] = -1;
  if (j == 0) { table[NN] = -1; smax = -2147483647; }
  __syncthreads();
  atomicMax(&smax, as);
  __syncthreads();
  const int mx = smax;
  if (as == mx && a > 0 && a <= NN) table[a] = j;   // product side: amn -> pos
  __syncthreads();
  const int m = (as < mx) ? a : 0;                  // reactant side amn
  pp[i * NN + j] = (m > 0 && m <= NN) ? table[m] : -1;
}

// ---------------------------------------------------------------------------
// Kernel 2: node features (X_cat, C_cat)
// ---------------------------------------------------------------------------
__global__ __launch_bounds__(256) void node_kernel(const float* __restrict__ X,
                                                   const float* __restrict__ C,
                                                   const int* __restrict__ pp,
                                                   float* __restrict__ out) {
  const int t = blockIdx.x * 256 + threadIdx.x;
  if (t >= NB * NN) return;
  const int i   = t / NN;
  const int p   = pp[t];
  const float m = (p >= 0) ? 1.0f : 0.0f;
  const int src = (p >= 0) ? p : 0;

  const float4* xr = (const float4*)(X + (long long)t * 16);
  const float4* xg = (const float4*)(X + ((long long)i * NN + src) * 16);
  float4* xo = (float4*)(out + XCAT_OFF + (long long)t * 32);
#pragma unroll
  for (int k = 0; k < 4; ++k) xo[k] = xr[k];
#pragma unroll
  for (int k = 0; k < 4; ++k) {
    float4 v = xg[k];
    v.x *= m; v.y *= m; v.z *= m; v.w *= m;
    xo[4 + k] = v;
  }
  const float* cr = C + (long long)t * 3;
  const float* cg = C + ((long long)i * NN + src) * 3;
  float* co = out + CCAT_OFF + (long long)t * 6;
  co[0] = cr[0];     co[1] = cr[1];     co[2] = cr[2];
  co[3] = cg[0] * m; co[4] = cg[1] * m; co[5] = cg[2] * m;
}

// ---------------------------------------------------------------------------
// TDM: 1-D global -> LDS DMA of n8 8-byte elements (D# per ISA ch.8).
// ---------------------------------------------------------------------------
#if USE_TDM
__device__ __forceinline__ void tdm_load_1d_u64(const void* gsrc, void* lds_dst, u32 n8) {
  const uint64_t ga = (uint64_t)(uintptr_t)gsrc;
  const u32 lo  = (u32)__builtin_amdgcn_readfirstlane((int)(u32)ga);
  const u32 hi  = (u32)__builtin_amdgcn_readfirstlane((int)(u32)(ga >> 32));
  const u32 lds = (u32)__builtin_amdgcn_readfirstlane((int)(u32)(uintptr_t)lds_dst);

  v4u g0;                                     // D# group 0 (128b)
  g0.x = 1u;                                  // count=1, is_restore=0, gather=0
  g0.y = lds;                                 // lds_addr [63:32]
  g0.z = lo;                                  // global_addr[31:0]
  g0.w = (hi & 0x01FFFFFFu) | (2u << 30);     // global_addr[56:32], type=2 (image)

  v8i g1;                                     // D# group 1 (256b)
  g1[0] = (int)(3u << 16);                    // workgroup_mask=0, data_size=3 (8B)
  g1[1] = (int)((n8 & 0xFFFFu) << 16);        // tensor_dim0[15:0]  (bits 63:48)
  g1[2] = (int)(((n8 >> 16) & 0xFFFFu) |      // tensor_dim0[31:16]
                (1u << 16));                  // tensor_dim1 = 1
  g1[3] = (int)((n8 & 0xFFFFu) << 16);        // tile_dim0 = n8 (bits 127:112)
  g1[4] = 0;                                  // tile_dim1=0 (unused), tile_dim2=0
  g1[5] = (int)n8;                            // tensor_dim0_stride[31:0]
  g1[6] = 0;                                  // stride hi / dim1 stride
  g1[7] = 0;

  v4i g2 = {0, 0, 0, 0};
  v4i g3 = {0, 0, 0, 0};
#  if defined(__clang_major__) && __clang_major__ >= 23
  v8i g4 = {0, 0, 0, 0, 0, 0, 0, 0};
  __builtin_amdgcn_tensor_load_to_lds(g0, g1, g2, g3, g4, 0);
#  else
  __builtin_amdgcn_tensor_load_to_lds(g0, g1, g2, g3, 0);
#  endif
}
#endif

// ---------------------------------------------------------------------------
// Kernel 3: edge features. One block per (batch i, row j1). TDM stages the
// gather sources (second half of source row p[j1] of E and BD, plus the pp
// row) into LDS; threads then own 16B-aligned float4 chunks of the output
// rows, gather from LDS, and stream wide coalesced b128 stores.
// ---------------------------------------------------------------------------
__global__ __launch_bounds__(NN) void edge_kernel(const float* __restrict__ E,
                                                  const float* __restrict__ BD,
                                                  const int* __restrict__ pp,
                                                  float* __restrict__ out) {
  const int j1  = blockIdx.x;
  const int i   = blockIdx.y;
  const int tid = threadIdx.x;

  __shared__ double sE[NHALF * 5 / 2];   // 640 * 8B = 5 KB  (E[i,p1,256:512,:5])
  __shared__ double sB[NHALF * 3 / 2];   // 384 * 8B = 3 KB  (BD[i,p1,256:512,:3])
  __shared__ int    sP[NN];              // 2 KB             (pp[i,:])

  const long long row = (long long)i * NN + j1;
  float4* __restrict__ e1 = (float4*)(out + E1_OFF  + row * NN * 5);  // 640 chunks
  float4* __restrict__ e2 = (float4*)(out + E2_OFF  + row * NN * 5);  // 640 chunks
  float4* __restrict__ b1 = (float4*)(out + BD1_OFF + row * NN * 3);  // 384 chunks

  const int ppj1 = __builtin_amdgcn_readfirstlane(pp[i * NN + j1]);

  if (ppj1 < 0) {
    // pair mask == 0 for the whole row: E1=0, BD1=0, E2 = onehot(0) pattern
    const float4 z = make_float4(0.f, 0.f, 0.f, 0.f);
    for (int t = tid; t < NN * 5 / 4; t += NN) {
      const int f = t * 4;
      float4 v;
      v.x = ((f + 0) % 5 == 0) ? 1.f : 0.f;
      v.y = ((f + 1) % 5 == 0) ? 1.f : 0.f;
      v.z = ((f + 2) % 5 == 0) ? 1.f : 0.f;
      v.w = ((f + 3) % 5 == 0) ? 1.f : 0.f;
      e1[t] = z;
      e2[t] = v;
    }
    for (int t = tid; t < NN * 3 / 4; t += NN) b1[t] = z;
    return;
  }

  const double* esrc = (const double*)(E  + (((long long)i * NN + ppj1) * NN + NHALF) * 5);
  const double* bsrc = (const double*)(BD + (((long long)i * NN + ppj1) * NN + NHALF) * 3);
  const double* psrc = (const double*)(pp + i * NN);   // 512 ints, 8B aligned

#if USE_TDM
  if ((tid >> 5) == 0) {                   // wave 0 only issues the DMAs
    tdm_load_1d_u64(esrc, sE, (u32)(NHALF * 5 / 2));
    tdm_load_1d_u64(bsrc, sB, (u32)(NHALF * 3 / 2));
    tdm_load_1d_u64(psrc, sP, (u32)(NN / 2));
    __builtin_amdgcn_s_wait_tensorcnt(0);
  }
  __syncthreads();
#else
  for (int t = tid; t < NHALF * 5 / 2; t += NN) sE[t] = esrc[t];
  for (int t = tid; t < NHALF * 3 / 2; t += NN) sB[t] = bsrc[t];
  for (int t = tid; t < NN / 2; t += NN) ((double*)sP)[t] = psrc[t];
  __syncthreads();
#endif

  const float* fE = (const float*)sE;
  const float* fB = (const float*)sB;

  // E1 / E2: each thread owns float4 chunks of the 2560-float row
  for (int t = tid; t < NN * 5 / 4; t += NN) {
    float v[4], w[4];
#pragma unroll
    for (int e = 0; e < 4; ++e) {
      const int f  = t * 4 + e;
      const int j2 = f / 5;
      const int c  = f - 5 * j2;
      const int p  = sP[j2];
      v[e] = (p >= 0) ? fE[(p - NHALF) * 5 + c] : 0.f;   // product pos >= NHALF
      w[e] = (p < 0 && c == 0) ? 1.f : 0.f;
    }
    e1[t] = make_float4(v[0], v[1], v[2], v[3]);
    e2[t] = make_float4(w[0], w[1], w[2], w[3]);
  }

  // BD1: 1536-float row as 384 float4 chunks
  for (int t = tid; t < NN * 3 / 4; t += NN) {
    float v[4];
#pragma unroll
    for (int e = 0; e < 4; ++e) {
      const int f  = t * 4 + e;
      const int j2 = f / 3;
      const int c  = f - 3 * j2;
      const int p  = sP[j2];
      v[e] = (p >= 0) ? fB[(p - NHALF) * 3 + c] : 0.f;
    }
    b1[t] = make_float4(v[0], v[1], v[2], v[3]);
  }
}

// ---------------------------------------------------------------------------
extern "C" void kernel_launch(void* const* d_in, const int* in_sizes, int n_in,
                              void* d_out, int out_size, void* d_ws, size_t ws_size,
                              hipStream_t stream) {
  (void)in_sizes; (void)n_in; (void)out_size; (void)ws_size;
  const float* X   = (const float*)d_in[0];
  const float* E   = (const float*)d_in[1];
  const float* C   = (const float*)d_in[2];
  const float* BD  = (const float*)d_in[3];
  const int*   amn = (const int*)d_in[4];
  const int*   mol = (const int*)d_in[5];
  float* out = (float*)d_out;
  int*   pp  = (int*)d_ws;                  // NB*NN ints = 64 KB scratch

  build_pp_kernel<<<NB, NN, 0, stream>>>(amn, mol, pp);
  node_kernel<<<(NB * NN + 255) / 256, 256, 0, stream>>>(X, C, pp, out);
  edge_kernel<<<dim3(NN, NB), NN, 0, stream>>>(E, BD, pp, out);
}